// AUCMLoss_55628416418005
// MI455X (gfx1250) — compile-verified
//
#include <hip/hip_runtime.h>
#include <math.h>

// AUC-margin pairwise loss for MI455X (gfx1250, wave32).
// - Per-class pos/neg compaction in LDS (~20x work cut at 5% positive rate).
// - 16x16 pair tiles: diff matrix produced as a rank-2 outer product on the
//   WMMA pipe (V_WMMA_F32_16X16X4_F32).
// - Base-2 softplus on raw v_exp_f32/v_log_f32 (neg/abs folded into exp's
//   source modifiers); inputs pre-scaled by MARGIN*log2(e), ln2 once/class.
// - Lists zero-padded to tile boundary -> unconditional, software-pipelined
//   LDS operand loads (no EXEC juggling / per-tile dscnt stall before WMMA).
// - Full-tile fast path (no masking) + masked edge path, wave-uniform.

typedef __attribute__((ext_vector_type(2))) float v2f;
typedef __attribute__((ext_vector_type(8))) float v8f;

#define MARGIN_F   1.0f
#define LOG2E_F    1.44269504088896340736f
#define LN2_F      0.69314718055994530942f
#define C_CLASSES  128
#define B_MAX      1024
#define LIST_PAD   32                    // one tile of zero pad + prefetch tile
#define THREADS    256
#define NWAVES     (THREADS / 32)

__device__ __forceinline__ float softplus2(float z) {
    // returns softplus(z*ln2)/ln2 = max(z,0) + log2(1 + exp2(-|z|))
    float e = __builtin_amdgcn_exp2f(-__builtin_fabsf(z));  // neg+abs src mods
    return __builtin_fmaxf(z, 0.0f) + __builtin_amdgcn_logf(1.0f + e);
}

__global__ __launch_bounds__(THREADS)
void aucm_per_class_kernel(const float* __restrict__ logits,
                           const float* __restrict__ targets,
                           float* __restrict__ per_class,
                           float* __restrict__ validf,
                           int B)
{
    const int c   = blockIdx.x;      // one workgroup per class
    const int tid = threadIdx.x;

    __shared__ float         probs[B_MAX];
    __shared__ unsigned char isPos[B_MAX];
    __shared__ int           posCnt[THREADS], negCnt[THREADS];
    __shared__ int           posOff[THREADS], negOff[THREADS];
    __shared__ float         posA[B_MAX + LIST_PAD]; // -m*log2e*p, zero-padded
    __shared__ float         negB[B_MAX + LIST_PAD]; //  m*log2e*p, zero-padded
    __shared__ int           npnn[2];
    __shared__ float         waveSum[NWAVES];

    const int rowsPer = (B + THREADS - 1) / THREADS;   // 4 for B=1024

    // ---- Phase 1: sigmoid + classify (deterministic contiguous chunks) ----
    int myPos = 0, myNeg = 0;
    for (int r = 0; r < rowsPer; ++r) {
        int row = tid * rowsPer + r;
        if (row < B && row < B_MAX) {
            float x = logits[row * C_CLASSES + c];
            // sigmoid via base-2 hardware exp: 1/(1+2^(-x*log2e))
            float p = __builtin_amdgcn_rcpf(
                1.0f + __builtin_amdgcn_exp2f(-LOG2E_F * x));
            probs[row] = p;
            int pos = targets[row * C_CLASSES + c] > 0.5f;
            isPos[row] = (unsigned char)pos;
            myPos += pos;
            myNeg += 1 - pos;
        }
    }
    posCnt[tid] = myPos;
    negCnt[tid] = myNeg;
    __syncthreads();

    // ---- Phase 2: serial exclusive scan (deterministic) ----
    if (tid == 0) {
        int pa = 0, na = 0;
        for (int t = 0; t < THREADS; ++t) {
            posOff[t] = pa; negOff[t] = na;
            pa += posCnt[t]; na += negCnt[t];
        }
        npnn[0] = pa; npnn[1] = na;
    }
    __syncthreads();

    const int np  = npnn[0];
    const int nn  = npnn[1];
    const int nti = (np + 15) >> 4;
    const int ntj = (nn + 15) >> 4;

    // ---- Phase 3: ordered scatter + zero-pad tails to tile boundary ----
    {
        const float kA = -MARGIN_F * LOG2E_F;
        const float kB =  MARGIN_F * LOG2E_F;
        int po = posOff[tid], no = negOff[tid];
        for (int r = 0; r < rowsPer; ++r) {
            int row = tid * rowsPer + r;
            if (row < B && row < B_MAX) {
                float p = probs[row];
                if (isPos[row]) posA[po++] = kA * p;
                else            negB[no++] = kB * p;
            }
        }
        // zero tails: [np, nti*16+16) and [nn, ntj*16+16) — covers the
        // prefetch tile, so all operand loads below are unconditional.
        for (int idx = np + tid; idx < nti * 16 + 16; idx += THREADS)
            posA[idx] = 0.0f;
        for (int idx = nn + tid; idx < ntj * 16 + 16; idx += THREADS)
            negB[idx] = 0.0f;
    }
    __syncthreads();

    // ---- Phase 4: 16x16 pair tiles via WMMA rank-2 outer product ----
    // D[i][j] = posA_i * 1 + 1 * negB_j = -m*log2e*(p_i - q_j) = z
    const int lane    = tid & 31;
    const int wave    = tid >> 5;
    const int halfSel = lane >> 4;      // 0: lanes 0-15, 1: lanes 16-31
    const int lane15  = lane & 15;

    // Lane-constant operand components, hoisted out of all loops.
    const float aOne = halfSel ? 0.0f : 1.0f;   // A column K=1 (ones)
    const float bOne = halfSel ? 0.0f : 1.0f;   // B row    K=0 (ones)

    float acc = 0.0f;

    // Each wave owns a strip of j-tiles (ntj ~ 61 over 8 waves); B operand
    // built once per strip. All branches are wave-uniform -> EXEC all-ones
    // at every WMMA.
    for (int tj = wave; tj < ntj; tj += NWAVES) {
        const int   j0 = tj << 4;
        const float bv = negB[j0 + lane15];       // unconditional (padded)
        v2f b;
        b.x = bOne;
        b.y = halfSel ? 0.0f : bv;
        const bool  colFull  = (j0 + 16) <= nn;                  // uniform
        const float colValid = ((j0 + lane15) < nn) ? 1.0f : 0.0f;

        // software pipeline: prefetch A value for tile ti+1 past current work
        float avCur = posA[lane15];               // ti = 0
        for (int ti = 0; ti < nti; ++ti) {
            const int i0 = ti << 4;
            float avNext = posA[i0 + 16 + lane15];   // safe: padded tile

            v2f a;
            a.x = halfSel ? 0.0f : avCur;
            a.y = aOne;

            v8f c8 = {};
            v8f d = __builtin_amdgcn_wmma_f32_16x16x4_f32(
                false, a, false, b, (short)0, c8, false, false);

            // D layout: lanes 0-15 -> (M=k, N=lane); lanes 16-31 -> (M=k+8).
            if (colFull && (i0 + 16) <= np) {
                // fast path: whole tile valid, zero masking work
                #pragma unroll
                for (int k = 0; k < 8; ++k)
                    acc += softplus2(d[k]);
            } else {
                #pragma unroll
                for (int k = 0; k < 8; ++k) {
                    int   row = i0 + k + (halfSel << 3);
                    float w   = (row < np) ? colValid : 0.0f;
                    acc += w * softplus2(d[k]);
                }
            }
            avCur = avNext;
        }
    }

    // ---- Phase 5: deterministic reduction ----
    for (int off = 16; off > 0; off >>= 1)
        acc += __shfl_xor(acc, off, 32);
    if (lane == 0) waveSum[wave] = acc;
    __syncthreads();

    if (tid == 0) {
        float num = 0.0f;
        for (int w2 = 0; w2 < NWAVES; ++w2) num += waveSum[w2];
        num *= LN2_F;                        // undo base-2 scaling once
        float cnt   = (float)np * (float)nn;
        bool  valid = cnt > 0.0f;
        per_class[c] = valid ? (num / cnt) : 0.0f;
        validf[c]    = valid ? 1.0f : 0.0f;
    }
}

__global__ __launch_bounds__(C_CLASSES)
void aucm_finalize_kernel(const float* __restrict__ per_class,
                          const float* __restrict__ validf,
                          float* __restrict__ out)
{
    __shared__ float s1[C_CLASSES];
    __shared__ float s2[C_CLASSES];
    int t = threadIdx.x;
    s1[t] = per_class[t];
    s2[t] = validf[t];
    __syncthreads();
    for (int s = C_CLASSES / 2; s > 0; s >>= 1) {
        if (t < s) { s1[t] += s1[t + s]; s2[t] += s2[t + s]; }
        __syncthreads();
    }
    if (t == 0)
        out[0] = (s2[0] > 0.0f) ? (s1[0] / s2[0]) : 0.0f;
}

extern "C" void kernel_launch(void* const* d_in, const int* in_sizes, int n_in,
                              void* d_out, int out_size, void* d_ws, size_t ws_size,
                              hipStream_t stream)
{
    (void)n_in; (void)out_size; (void)ws_size;
    const float* logits  = (const float*)d_in[0];
    const float* targets = (const float*)d_in[1];
    float*       out     = (float*)d_out;

    const int B = in_sizes[0] / C_CLASSES;   // reference: B=1024, C=128

    float* per_class = (float*)d_ws;              // C floats
    float* validf    = per_class + C_CLASSES;     // C floats  (1 KB total)

    aucm_per_class_kernel<<<C_CLASSES, THREADS, 0, stream>>>(
        logits, targets, per_class, validf, B);
    aucm_finalize_kernel<<<1, C_CLASSES, 0, stream>>>(per_class, validf, out);
}